// Voxelization_50345606644201
// MI455X (gfx1250) — compile-verified
//
#include <hip/hip_runtime.h>
#include <stdint.h>

// ---- problem constants (match reference) ----
#define GX 1408
#define GY 1600
#define GZ 40
#define GYGX 2252800            // GX*GY
#define MAXV 120000
#define MAXP 35
#define FEAT 4
#define SCAN_TILE 2048          // 256 threads * 8 items
#define IMAXI 0x7FFFFFFF

// ---------------- device helpers ----------------
__device__ __forceinline__ uint32_t hash_slot(uint32_t key, int logc) {
    return (key * 2654435761u) >> (32 - logc);
}

// returns true if point is in range; sets lin (linear voxel id)
__device__ __forceinline__ bool point_cell(const float4& q, int& lin) {
    // replicate: floor((p - lo) / vsz)  in f32, then bounds check
    float fx = floorf((q.x - 0.0f)  / 0.05f);
    float fy = floorf((q.y + 40.0f) / 0.05f);
    float fz = floorf((q.z + 3.0f)  / 0.1f);
    bool valid = (fx >= 0.0f) && (fx < (float)GX) &&
                 (fy >= 0.0f) && (fy < (float)GY) &&
                 (fz >= 0.0f) && (fz < (float)GZ);
    if (!valid) return false;
    int cx = (int)fx, cy = (int)fy, cz = (int)fz;
    lin = cz * GYGX + cy * GX + cx;
    return true;
}

// ---------------- init kernels ----------------
__global__ void kZero32(int* p, long long n) {
    long long i = (long long)blockIdx.x * blockDim.x + threadIdx.x;
    long long s = (long long)gridDim.x * blockDim.x;
    for (; i < n; i += s) p[i] = 0;
}

__global__ void kFill32(int* p, int v, int n) {
    int i = blockIdx.x * blockDim.x + threadIdx.x;
    if (i < n) p[i] = v;
}

__global__ void kInitHash(int* hkey, int* hmin, int C) {
    int i = blockIdx.x * blockDim.x + threadIdx.x;
    if (i < C) { hkey[i] = -1; hmin[i] = IMAXI; }
}

// ---------------- pass A: hash insert + first-occurrence index ----------------
__global__ void kInsert(const float4* __restrict__ pts, int n,
                        int* hkey, int* hmin, int C, int logc) {
    int i = blockIdx.x * blockDim.x + threadIdx.x;
    if (i >= n) return;
    float4 q = pts[i];
    int lin;
    if (!point_cell(q, lin)) return;
    uint32_t mask = (uint32_t)C - 1u;
    uint32_t s = hash_slot((uint32_t)lin, logc) & mask;
    for (;;) {
        int prev = atomicCAS(&hkey[s], -1, lin);
        if (prev == -1 || prev == lin) { atomicMin(&hmin[s], i); break; }
        s = (s + 1u) & mask;
    }
}

// ---------------- pass B1: mark first-occurrence point indices ----------------
__global__ void kMark(const int* __restrict__ hkey, const int* __restrict__ hmin,
                      int C, int* __restrict__ mark) {
    int i = blockIdx.x * blockDim.x + threadIdx.x;
    if (i >= C) return;
    if (hkey[i] < 0) return;
    mark[hmin[i]] = 1;
}

// ---------------- scan: per-block exclusive prefix (in place) ----------------
__global__ void kScanBlocks(int* __restrict__ data, int n, int* __restrict__ blockSums) {
    __shared__ int sh[256];
    int base = blockIdx.x * SCAN_TILE + threadIdx.x * 8;
    int v[8]; int sum = 0;
    #pragma unroll
    for (int k = 0; k < 8; k++) {
        int idx = base + k;
        v[k] = (idx < n) ? data[idx] : 0;
        sum += v[k];
    }
    sh[threadIdx.x] = sum;
    __syncthreads();
    #pragma unroll
    for (int off = 1; off < 256; off <<= 1) {
        int t = (threadIdx.x >= off) ? sh[threadIdx.x - off] : 0;
        __syncthreads();
        sh[threadIdx.x] += t;
        __syncthreads();
    }
    int excl = sh[threadIdx.x] - sum;
    if (threadIdx.x == 255) blockSums[blockIdx.x] = sh[255];
    int run = excl;
    #pragma unroll
    for (int k = 0; k < 8; k++) {
        int idx = base + k;
        if (idx < n) { int t = v[k]; data[idx] = run; run += t; }
    }
}

__global__ void kScanBlockSums(int* bs, int nb, int* totalOut) {
    if (blockIdx.x == 0 && threadIdx.x == 0) {
        int run = 0;
        for (int i = 0; i < nb; i++) { int t = bs[i]; bs[i] = run; run += t; }
        *totalOut = run;
    }
}

// ---------------- pass B2: assign voxel ids by first-occurrence rank ----------------
__global__ void kAssignVox(const int* __restrict__ hkey, int* __restrict__ hmin, int C,
                           const int* __restrict__ excl, const int* __restrict__ blockOff,
                           int* __restrict__ coors) {
    int i = blockIdx.x * blockDim.x + threadIdx.x;
    if (i >= C) return;
    int key = hkey[i];
    if (key < 0) return;
    int m = hmin[i];
    int rank = excl[m] + blockOff[m >> 11];     // SCAN_TILE == 2048
    if (rank < MAXV) {
        hmin[i] = rank;                         // entry now holds voxel id
        int cz = key / GYGX;
        int rem = key - cz * GYGX;
        int cy = rem / GX;
        int cx = rem - cy * GX;
        coors[rank * 3 + 0] = cz;               // reference stores (z, y, x)
        coors[rank * 3 + 1] = cy;
        coors[rank * 3 + 2] = cx;
    } else {
        hmin[i] = -1;                           // dropped voxel
    }
}

// ---------------- pass C: compact points that land in kept voxels ----------------
__global__ void kCompact(const float4* __restrict__ pts, int n,
                         const int* __restrict__ hkey, const int* __restrict__ hmin,
                         int C, int logc, int* cnt,
                         int* __restrict__ cIdx, int* __restrict__ cVox) {
    int i = blockIdx.x * blockDim.x + threadIdx.x;
    if (i >= n) return;
    float4 q = pts[i];
    int lin;
    if (!point_cell(q, lin)) return;
    uint32_t mask = (uint32_t)C - 1u;
    uint32_t s = hash_slot((uint32_t)lin, logc) & mask;
    while (hkey[s] != lin) s = (s + 1u) & mask;   // key guaranteed present
    int vox = hmin[s];
    if (vox < 0) return;
    int pos = atomicAdd(cnt, 1);
    cIdx[pos] = i;
    cVox[pos] = vox;
}

// ---------------- 35 deterministic selection rounds ----------------
__global__ void kRoundMin(const int* __restrict__ cnt,
                          const int* __restrict__ cIdx, const int* __restrict__ cVox,
                          int* __restrict__ cur) {
    int i = blockIdx.x * blockDim.x + threadIdx.x;
    if (i >= *cnt) return;
    int p = cIdx[i];
    if (p < 0) return;
    atomicMin(&cur[cVox[i]], p);
}

__global__ void kRoundResolve(const int* __restrict__ cnt,
                              int* __restrict__ cIdx, const int* __restrict__ cVox,
                              int* __restrict__ cur,
                              const float4* __restrict__ pts,
                              float* __restrict__ voxels, int* __restrict__ npts, int r) {
    int i = blockIdx.x * blockDim.x + threadIdx.x;
    if (i >= *cnt) return;
    int p = cIdx[i];
    if (p < 0) return;
    int v = cVox[i];
    if (cur[v] == p) {                  // unique winner: smallest remaining index in voxel v
        float4 q = pts[p];
        float* dst = voxels + ((size_t)v * MAXP + r) * FEAT;   // 16B aligned
        *(float4*)dst = q;
        npts[v] = r + 1;
        cIdx[i] = -1;                   // retire point
        cur[v] = IMAXI;                 // reset for next round
    }
}

__global__ void kFinal(const int* total, int* voxnum) {
    int t = *total;
    *voxnum = (t < MAXV) ? t : MAXV;
}

// ---------------- host launcher ----------------
extern "C" void kernel_launch(void* const* d_in, const int* in_sizes, int n_in,
                              void* d_out, int out_size, void* d_ws, size_t ws_size,
                              hipStream_t stream) {
    (void)n_in;
    const float4* pts = (const float4*)d_in[0];
    const int N = in_sizes[0] / 4;             // points: [N, 4]

    float* voxels = (float*)d_out;                                  // [MAXV,35,4] f32
    int* coors    = (int*)((float*)d_out + (size_t)MAXV * MAXP * FEAT); // [MAXV,3] i32
    int* npts     = coors + (size_t)MAXV * 3;                       // [MAXV] i32
    int* voxnum   = npts + MAXV;                                    // [1] i32

    const int nb = (N + SCAN_TILE - 1) / SCAN_TILE;
    const int nbAlloc = (nb + 63) & ~63;

    // choose hash capacity: prefer 2^22 (load ~26%), fall back to 2^21 if ws is tight
    int lc = 22;
    {
        size_t need22 = ((size_t)1 << 23) * 4 + (size_t)N * 12 + (size_t)nbAlloc * 4
                      + (size_t)MAXV * 4 + 256;
        if (need22 > ws_size) lc = 21;
    }
    const int C = 1 << lc;

    char* w = (char*)d_ws;
    int* hkey      = (int*)w; w += (size_t)C * 4;
    int* hmin      = (int*)w; w += (size_t)C * 4;
    int* mark      = (int*)w; w += (size_t)N * 4;   // becomes exclusive prefix in place
    int* blockSums = (int*)w; w += (size_t)nbAlloc * 4;
    int* cIdx      = (int*)w; w += (size_t)N * 4;
    int* cVox      = (int*)w; w += (size_t)N * 4;
    int* cur       = (int*)w; w += (size_t)MAXV * 4;
    int* counters  = (int*)w;                       // [0]=compact count, [1]=group total

    const int B = 256;
    const int gN = (N + B - 1) / B;
    const int gC = (C + B - 1) / B;

    // init (fresh every call — graph-replay deterministic)
    kZero32<<<1024, B, 0, stream>>>((int*)d_out, (long long)out_size);
    kInitHash<<<gC, B, 0, stream>>>(hkey, hmin, C);
    kZero32<<<1024, B, 0, stream>>>(mark, (long long)N);
    kFill32<<<(MAXV + B - 1) / B, B, 0, stream>>>(cur, IMAXI, MAXV);
    kZero32<<<1, 32, 0, stream>>>(counters, 8);

    // A: first-occurrence index per occupied cell
    kInsert<<<gN, B, 0, stream>>>(pts, N, hkey, hmin, C, lc);
    // B: rank cells by first occurrence via prefix scan over point indices
    kMark<<<gC, B, 0, stream>>>(hkey, hmin, C, mark);
    kScanBlocks<<<nb, 256, 0, stream>>>(mark, N, blockSums);
    kScanBlockSums<<<1, 1, 0, stream>>>(blockSums, nb, counters + 1);
    kAssignVox<<<gC, B, 0, stream>>>(hkey, hmin, C, mark, blockSums, coors);
    // C: compact points in kept voxels
    kCompact<<<gN, B, 0, stream>>>(pts, N, hkey, hmin, C, lc, counters + 0, cIdx, cVox);
    // D: 35 exact selection rounds (slot r = (r+1)-th smallest original index in voxel)
    for (int r = 0; r < MAXP; r++) {
        kRoundMin<<<gN, B, 0, stream>>>(counters + 0, cIdx, cVox, cur);
        kRoundResolve<<<gN, B, 0, stream>>>(counters + 0, cIdx, cVox, cur,
                                            pts, voxels, npts, r);
    }
    kFinal<<<1, 1, 0, stream>>>(counters + 1, voxnum);
}